// DecoderRNN_56762287784616
// MI455X (gfx1250) — compile-verified
//
#include <hip/hip_runtime.h>
#include <math.h>

#define H 1024
#define V 50257
#define L 40

typedef float f4  __attribute__((ext_vector_type(4)));
typedef float v2f __attribute__((ext_vector_type(2)));
typedef float v8f __attribute__((ext_vector_type(8)));
typedef unsigned int u32x4 __attribute__((ext_vector_type(4)));
typedef int i32x4 __attribute__((ext_vector_type(4)));
typedef int i32x8 __attribute__((ext_vector_type(8)));

__device__ __forceinline__ float waveReduceSum(float v) {
#pragma unroll
  for (int off = 16; off > 0; off >>= 1)
    v += __shfl_xor(v, off, 32);
  return v;
}

// ---- TDM: 1-D tensor DMA of `nelem` f32 from global -> LDS byte offset ----
// D# per CDNA5 ISA ch.8: group0 = {count, lds_addr, global_addr[56:0], type=2},
// group1 = {data_size=4B, tensor_dim0, tile_dim0, tensor_dim0_stride}.
#if __has_builtin(__builtin_amdgcn_tensor_load_to_lds)
#define HAVE_TDM 1
__device__ __forceinline__ void tdm_load_1d(const float* gsrc, unsigned lds_byte_off,
                                            int nelem) {
  unsigned long long ga = (unsigned long long)(uintptr_t)gsrc;
  u32x4 g0;
  g0.x = 1u;                                              // count=1 (valid user D#)
  g0.y = lds_byte_off;                                    // lds_addr [63:32]
  g0.z = (unsigned)ga;                                    // global_addr [95:64]
  g0.w = (unsigned)((ga >> 32) & 0x01FFFFFFu) | (2u << 30); // addr[56:32] | type=2
  i32x8 g1;
  g1[0] = (2 << 16);                                      // data_size=2 (4 bytes)
  g1[1] = (int)(((unsigned)nelem & 0xFFFFu) << 16);       // tensor_dim0[15:0] @ bits48-63
  g1[2] = (int)((((unsigned)nelem >> 16) & 0xFFFFu) | (1u << 16)); // dim0 hi | tensor_dim1=1
  g1[3] = (int)(((unsigned)nelem & 0xFFFFu) << 16);       // tile_dim0 @ bits112-127
  g1[4] = 0;                                              // tile_dim1=0, tile_dim2=0
  g1[5] = nelem;                                          // tensor_dim0_stride low32
  g1[6] = 0;
  g1[7] = 0;
  i32x4 z4 = {0, 0, 0, 0};
#if __clang_major__ >= 23
  i32x8 z8 = {0, 0, 0, 0, 0, 0, 0, 0};
  __builtin_amdgcn_tensor_load_to_lds(g0, g1, z4, z4, z8, 0);
#else
  __builtin_amdgcn_tensor_load_to_lds(g0, g1, z4, z4, 0);
#endif
}
#else
#define HAVE_TDM 0
#endif

// ---------------- K1: x = relu(embedding[token]) ----------------
__global__ void k_embed(const int* __restrict__ tok, const float* __restrict__ emb,
                        float* __restrict__ x) {
  int i = blockIdx.x * blockDim.x + threadIdx.x;
  if (i < H) {
    float v = emb[(size_t)tok[0] * H + i];
    x[i] = v > 0.f ? v : 0.f;
  }
}

// ---------------- K2: gates = W_ih@x + W_hh@h + b  (4096-row GEMV) ----------------
// x,h staged to LDS via Tensor Data Mover (wave0 issues, waits TENSORcnt, barrier);
// one row per wave32 with coalesced b128 loads.
__global__ void k_gates(const float* __restrict__ Wih, const float* __restrict__ Whh,
                        const float* __restrict__ bih, const float* __restrict__ bhh,
                        const float* __restrict__ x, const float* __restrict__ h,
                        float* __restrict__ gates) {
  __shared__ float xs[H];   // LDS offset 0
  __shared__ float hs[H];   // LDS offset 4096
  int tid = threadIdx.x;
#if HAVE_TDM
  if (tid < 32) {                       // wave 0 issues both DMAs (TDM ignores EXEC)
    tdm_load_1d(x, 0u, H);
    tdm_load_1d(h, (unsigned)(H * sizeof(float)), H);
    __builtin_amdgcn_s_wait_tensorcnt(0);
  }
#else
  for (int i = tid; i < H / 4; i += blockDim.x) {
    ((f4*)xs)[i] = ((const f4*)x)[i];
    ((f4*)hs)[i] = ((const f4*)h)[i];
  }
#endif
  __syncthreads();
  int wave = tid >> 5, lane = tid & 31;
  int row = blockIdx.x * 8 + wave;      // grid 512 * 8 = 4096 rows exactly
  const f4* wi = (const f4*)(Wih + (size_t)row * H);
  const f4* wh = (const f4*)(Whh + (size_t)row * H);
  const f4* x4 = (const f4*)xs;
  const f4* h4 = (const f4*)hs;
  float acc = 0.f;
#pragma unroll
  for (int i = 0; i < 8; ++i) {
    int c = lane + i * 32;
    f4 a = wi[c], b = wh[c];
    f4 xa = x4[c], hb = h4[c];
    acc += a.x * xa.x + a.y * xa.y + a.z * xa.z + a.w * xa.w;
    acc += b.x * hb.x + b.y * hb.y + b.z * hb.z + b.w * hb.w;
  }
  acc = waveReduceSum(acc);
  if (lane == 0) gates[row] = acc + bih[row] + bhh[row];
}

// ---------------- K3: LSTM elementwise ----------------
__global__ void k_lstm(const float* __restrict__ gates, const float* __restrict__ c0,
                       float* __restrict__ hn_ws, float* __restrict__ hn_out,
                       float* __restrict__ cn_out) {
  int t = blockIdx.x * blockDim.x + threadIdx.x;
  if (t < H) {
    float ig = 1.f / (1.f + expf(-gates[t]));
    float fg = 1.f / (1.f + expf(-gates[H + t]));
    float gg = tanhf(gates[2 * H + t]);
    float og = 1.f / (1.f + expf(-gates[3 * H + t]));
    float cn = fg * c0[t] + ig * gg;
    float hv = og * tanhf(cn);
    hn_ws[t] = hv;
    hn_out[t] = hv;
    cn_out[t] = cn;
  }
}

// ---------------- K4: raw-dot attention + ctx + concat (single 1024-thread block) ---
__global__ void k_attn(const float* __restrict__ enc, const float* __restrict__ hn,
                       float* __restrict__ attn_out, float* __restrict__ vcat) {
  __shared__ float hs[H];
  __shared__ float aw[L];
  int tid = threadIdx.x;
  hs[tid] = hn[tid];
  __syncthreads();
  int wave = tid >> 5, lane = tid & 31;
  const f4* h4 = (const f4*)hs;
  for (int l = wave; l < L; l += 32) {
    const f4* e4 = (const f4*)(enc + (size_t)l * H);
    float acc = 0.f;
#pragma unroll
    for (int i = 0; i < 8; ++i) {
      int c = lane + i * 32;
      f4 a = e4[c], b = h4[c];
      acc += a.x * b.x + a.y * b.y + a.z * b.z + a.w * b.w;
    }
    acc = waveReduceSum(acc);
    if (lane == 0) { aw[l] = acc; attn_out[l] = acc; }
  }
  __syncthreads();
  float s = 0.f;
#pragma unroll 8
  for (int l = 0; l < L; ++l) s += aw[l] * enc[(size_t)l * H + tid];
  vcat[tid] = s;            // ctx
  vcat[H + tid] = hs[tid];  // h_new
}

// ---------------- K5: out_vec = tanh(W_ctx @ vcat + b_ctx), via WMMA f32 16x16x4 ----
// One wave per 16-row tile. A layout (ISA 7.12.2, 32-bit A 16x4):
//   VGPR0: lanes0-15 K=0 (M=lane), lanes16-31 K=2 (M=lane-16); VGPR1: K=1 / K=3.
// B = vcat broadcast across all 16 N columns -> D[m][n] == y[m] for every n.
__global__ void k_outvec_wmma(const float* __restrict__ Wctx, const float* __restrict__ bctx,
                              const float* __restrict__ vcat, float* __restrict__ ovec) {
  __shared__ float vs[2 * H];
  int tid = threadIdx.x;
  for (int i = tid; i < (2 * H) / 4; i += blockDim.x)
    ((f4*)vs)[i] = ((const f4*)vcat)[i];
  __syncthreads();
  int wave = tid >> 5, lane = tid & 31;
  int m0 = (blockIdx.x * 8 + wave) * 16;   // 8 blocks * 8 waves = 64 tiles = 1024 rows
  int r  = m0 + (lane & 15);
  int cg = (lane >> 4) << 1;               // 0 for lanes 0-15, 2 for lanes 16-31
  const float* Wr = Wctx + (size_t)r * (2 * H) + cg;
  const float* vp = vs + cg;
  v8f acc = {0.f, 0.f, 0.f, 0.f, 0.f, 0.f, 0.f, 0.f};
  for (int kb = 0; kb < 2 * H; kb += 4) {
    v2f a = *(const v2f*)(Wr + kb);        // A[m][kb+cg], A[m][kb+cg+1]
    v2f b = *(const v2f*)(vp + kb);        // B[k][n] = vcat[kb+k] (broadcast over n)
    acc = __builtin_amdgcn_wmma_f32_16x16x4_f32(
        /*neg_a=*/false, a, /*neg_b=*/false, b,
        /*c_mod=*/(short)0, acc, /*reuse_a=*/false, /*reuse_b=*/false);
  }
  // D layout: acc[d] lanes0-15 -> y[m0+d]; lanes16-31 -> y[m0+8+d]
  if (lane == 0 || lane == 16) {
    int base = m0 + ((lane >> 4) << 3);
#pragma unroll
    for (int d = 0; d < 8; ++d)
      ovec[base + d] = tanhf(acc[d] + bctx[base + d]);
  }
}

// ---------------- K6: logits = W_out @ out_vec + b_out (50257-row GEMV, NT loads) ----
__global__ void k_logits(const float* __restrict__ Wout, const float* __restrict__ bout,
                         const float* __restrict__ ovec, float* __restrict__ logits) {
  __shared__ float os[H];
  int tid = threadIdx.x;
  for (int i = tid; i < H / 4; i += blockDim.x)
    ((f4*)os)[i] = ((const f4*)ovec)[i];
  __syncthreads();
  int wave = tid >> 5, lane = tid & 31;
  int row = blockIdx.x * 8 + wave;
  if (row >= V) return;
  const f4* w4 = (const f4*)(Wout + (size_t)row * H);
  const f4* o4 = (const f4*)os;
  float acc = 0.f;
#pragma unroll
  for (int i = 0; i < 8; ++i) {
    int c = lane + i * 32;
    f4 a = __builtin_nontemporal_load(&w4[c]);   // TH=NT: don't thrash 192MB L2
    f4 b = o4[c];
    acc += a.x * b.x + a.y * b.y + a.z * b.z + a.w * b.w;
  }
  acc = waveReduceSum(acc);
  if (lane == 0) logits[row] = acc + bout[row];
}

// ---------------- K7a: per-block online log-sum-exp partials ----------------
__global__ void k_lse_partial(const float* __restrict__ logits, float* __restrict__ pm,
                              float* __restrict__ ps) {
  __shared__ float sm[256], ss[256];
  int b = blockIdx.x, tid = threadIdx.x;
  const int chunk = (V + 63) / 64;                 // 786 (>=256 -> every thread gets work)
  int s0 = b * chunk;
  int e = s0 + chunk; if (e > V) e = V;
  float m = -INFINITY, s = 0.f;
  for (int i = s0 + tid; i < e; i += 256) {
    float v = logits[i];
    if (v > m) { s = s * expf(m - v) + 1.f; m = v; }
    else        s += expf(v - m);
  }
  sm[tid] = m; ss[tid] = s;
  __syncthreads();
  for (int off = 128; off > 0; off >>= 1) {
    if (tid < off) {
      float m2 = sm[tid + off], s2 = ss[tid + off];
      float M = fmaxf(sm[tid], m2);
      ss[tid] = ss[tid] * expf(sm[tid] - M) + s2 * expf(m2 - M);
      sm[tid] = M;
    }
    __syncthreads();
  }
  if (tid == 0) { pm[b] = sm[0]; ps[b] = ss[0]; }
}

// ---------------- K7b: combine 64 partials -> L = max + log(sum) ----------------
__global__ void k_lse_combine(const float* __restrict__ pm, const float* __restrict__ ps,
                              float* __restrict__ Lout) {
  int lane = threadIdx.x;   // single wave32
  float m1 = pm[lane], s1 = ps[lane];
  float m2 = pm[lane + 32], s2 = ps[lane + 32];
  float M = fmaxf(m1, m2);
  float S = s1 * expf(m1 - M) + s2 * expf(m2 - M);
#pragma unroll
  for (int off = 16; off > 0; off >>= 1) {
    float Mo = __shfl_xor(M, off, 32);
    float So = __shfl_xor(S, off, 32);
    float Mn = fmaxf(M, Mo);
    S = S * expf(M - Mn) + So * expf(Mo - Mn);
    M = Mn;
  }
  if (lane == 0) Lout[0] = M + logf(S);
}

// ---------------- K7c: logp = logits - L (in place) ----------------
__global__ void k_apply(float* __restrict__ logp, const float* __restrict__ Lv) {
  int i = blockIdx.x * blockDim.x + threadIdx.x;
  if (i < V) logp[i] -= Lv[0];
}

extern "C" void kernel_launch(void* const* d_in, const int* in_sizes, int n_in,
                              void* d_out, int out_size, void* d_ws, size_t ws_size,
                              hipStream_t stream) {
  const int*   tok  = (const int*)d_in[0];
  const float* h0   = (const float*)d_in[1];
  const float* c0   = (const float*)d_in[2];
  const float* enc  = (const float*)d_in[3];
  const float* emb  = (const float*)d_in[4];
  const float* Wih  = (const float*)d_in[5];
  const float* Whh  = (const float*)d_in[6];
  const float* bih  = (const float*)d_in[7];
  const float* bhh  = (const float*)d_in[8];
  const float* Wctx = (const float*)d_in[9];
  const float* bctx = (const float*)d_in[10];
  const float* Wout = (const float*)d_in[11];
  const float* bout = (const float*)d_in[12];

  float* out    = (float*)d_out;
  float* logp   = out;               // V
  float* hn_out = out + V;           // H
  float* cn_out = out + V + H;       // H
  float* aw_out = out + V + 2 * H;   // L

  float* ws    = (float*)d_ws;
  float* x     = ws;          // H
  float* hn    = ws + 1024;   // H
  float* gates = ws + 2048;   // 4H
  float* vcat  = ws + 6144;   // 2H
  float* ovec  = ws + 8192;   // H
  float* pm    = ws + 9216;   // 64
  float* ps    = ws + 9280;   // 64
  float* Lv    = ws + 9344;   // 1

  k_embed<<<4, 256, 0, stream>>>(tok, emb, x);
  k_gates<<<512, 256, 0, stream>>>(Wih, Whh, bih, bhh, x, h0, gates);
  k_lstm<<<4, 256, 0, stream>>>(gates, c0, hn, hn_out, cn_out);
  k_attn<<<1, 1024, 0, stream>>>(enc, hn, aw_out, vcat);
  k_outvec_wmma<<<8, 256, 0, stream>>>(Wctx, bctx, vcat, ovec);
  k_logits<<<(V + 7) / 8, 256, 0, stream>>>(Wout, bout, ovec, logp);
  k_lse_partial<<<64, 256, 0, stream>>>(logp, pm, ps);
  k_lse_combine<<<1, 32, 0, stream>>>(pm, ps, Lv);
  k_apply<<<(V + 255) / 256, 256, 0, stream>>>(logp, Lv);
}